// Maltes_partial_forces_loss_71897752535121
// MI455X (gfx1250) — compile-verified
//
#include <hip/hip_runtime.h>
#include <math.h>

// ---------------------------------------------------------------------------
// pairwise partial-forces loss, MI455X (gfx1250, wave32)
// 75 MB f32 stream (memory-bound target ~3 us @ 23.3 TB/s) -> lean VALU via
// raw v_sqrt_f32 / v_rcp_f32; pairwise-distance Gram tiles on the matrix unit
// via V_WMMA_F32_16X16X4_F32; tile-constant addressing folded to immediates.
// ---------------------------------------------------------------------------

typedef __attribute__((ext_vector_type(2))) float v2f;
typedef __attribute__((ext_vector_type(8))) float v8f;

#ifndef __has_builtin
#define __has_builtin(x) 0
#endif
#if __has_builtin(__builtin_amdgcn_wmma_f32_16x16x4_f32)
#define HAVE_WMMA_F32X4 1
#else
#define HAVE_WMMA_F32X4 0
#endif

#define NB     256
#define MAXN   192
#define SPLIT  4           // blocks per molecule (grid.y)

__device__ __forceinline__ float fast_sqrt(float x) {
  return __builtin_amdgcn_sqrtf(x);   // v_sqrt_f32, ~1 ulp; inputs well-scaled
}
__device__ __forceinline__ float fast_rcp(float x) {
  return __builtin_amdgcn_rcpf(x);    // v_rcp_f32, ~1 ulp; inputs >= 1e-18
}

__global__ __launch_bounds__(256) void pf_loss_per_mol(
    const float* __restrict__ pf,      // [TOTAL, MAXN, 3]
    const float* __restrict__ pos,     // [TOTAL, 3]
    const int*   __restrict__ n_atoms, // [NB]
    double*      __restrict__ partial) // [SPLIT, NB]
{
  __shared__ float4 sp[MAXN];   // x, y, z, |p|^2
  __shared__ int    s_off;
  __shared__ double red[256];

  const int b     = blockIdx.x;
  const int slice = blockIdx.y;
  const int tid   = threadIdx.x;

  // ragged row offset of molecule b (prefix sum of n_atoms)
  if (tid == 0) {
    int o = 0;
    for (int k = 0; k < b; ++k) o += n_atoms[k];
    s_off = o;
  }
  __syncthreads();
  const int n   = n_atoms[b];
  const int off = s_off;

  // stage positions (+ squared norms) in LDS, zero-padded to MAXN
  if (tid < MAXN) {
    float4 v = make_float4(0.f, 0.f, 0.f, 0.f);
    if (tid < n) {
      const float* p = pos + (size_t)(off + tid) * 3;
      const float x = p[0], y = p[1], z = p[2];
      v = make_float4(x, y, z, x * x + y * y + z * z);
    }
    sp[tid] = v;
  }
  __syncthreads();

  const int nt     = (n + 15) >> 4;  // 16x16 tiles per dimension
  const int ntt    = nt * nt;
  const int wave   = tid >> 5;       // 8 waves / block
  const int lane   = tid & 31;
  const int laneLo = lane & 15;
  const int laneHi = lane >> 4;      // half-wave: K {0,1} vs {2,3}; M bias 0/8

  const size_t rowStride = (size_t)MAXN * 3;   // floats per partial_forces row

  float t1 = 0.f, t2 = 0.f, t3 = 0.f;

  for (int t = slice * 8 + wave; t < ntt; t += 8 * SPLIT) { // uniform per wave
    const int I0 = (t / nt) << 4;
    const int J0 = (t % nt) << 4;

    const int    j      = J0 + laneLo;    // <= 191, always in-bounds of sp
    const float4 pj     = sp[j];
    const bool   jvalid = (j < n);
    const int    ibase  = I0 + (laneHi << 3);

    // tile-constant base pointers; per-v offsets are compile-time immediates
    const float* rowI = pf + (size_t)(off + ibase) * rowStride + (size_t)j * 3;
    const float* rowJ = pf + (size_t)(off + j) * rowStride + (size_t)ibase * 3;

#if HAVE_WMMA_F32X4
    // A (16x4, MxK): lane holds row M=laneLo, K = 2*laneHi + {0,1}; K=3 is 0.
    // B (4x16, KxN): lane holds col N=laneLo, same K split.
    const float4 pa = sp[I0 + laneLo];
    v2f a, bb;
    a[0]  = laneHi ? pa.z : pa.x;
    a[1]  = laneHi ? 0.f  : pa.y;
    bb[0] = laneHi ? pj.z : pj.x;
    bb[1] = laneHi ? 0.f  : pj.y;
    v8f g = {0.f, 0.f, 0.f, 0.f, 0.f, 0.f, 0.f, 0.f};
    // g[v] = dot(pos[I0 + v + 8*laneHi], pos[J0 + laneLo])
    g = __builtin_amdgcn_wmma_f32_16x16x4_f32(false, a, false, bb, (short)0, g,
                                              false, false);
#endif

#pragma unroll
    for (int v = 0; v < 8; ++v) {
      const int i = ibase + v;                // C/D layout: M = v + 8*laneHi
      if (!jvalid || i >= n) continue;
      const float4 pi = sp[i];

      const float rx = pi.x - pj.x, ry = pi.y - pj.y, rz = pi.z - pj.z;
#if HAVE_WMMA_F32X4
      const float d2 = fmaxf(pi.w + pj.w - 2.0f * g[v], 0.0f);
#else
      const float d2 = rx * rx + ry * ry + rz * rz;
#endif
      const float D     = (i == j) ? 0.0f : fast_sqrt(d2); // exact 0 on diag
      const float invDe = fast_rcp(D + 1e-3f);

      const float* Fp = rowI + (size_t)v * rowStride;  // P[i,j]
      const float* Gp = rowJ + v * 3;                  // P[j,i]
      const float fx = Fp[0], fy = Fp[1], fz = Fp[2];
      const float gx = Gp[0], gy = Gp[1], gz = Gp[2];

      const float fn2 = fx * fx + fy * fy + fz * fz;
      const float gn2 = gx * gx + gy * gy + gz * gz;
      const float pnF = fast_sqrt(fn2);
      const float pnG = fast_sqrt(gn2);

      // term1: cos(P_ij, P_ji) / (D + 1e-3)
      const float dotFG  = fx * gx + fy * gy + fz * gz;
      const float cos_pp = dotFG * fast_rcp(fmaxf(pnF * pnG, 1e-18f));
      t1 += cos_pp * invDe;

      // term2: (||P_ij|| - ||P_ji||)^2
      const float dn = pnF - pnG;
      t2 += dn * dn;

      // term3: -|cos(P_ij, r_ij)| / (D + 1e-3), off-diagonal only
      if (i != j) {
        const float dotFR  = fx * rx + fy * ry + fz * rz;
        const float cos_pr = dotFR * fast_rcp(fmaxf(pnF * D, 1e-18f));
        t3 -= fabsf(cos_pr) * invDe;
      }
    }
  }

  // per-molecule weights are uniform -> combine per lane, reduce in double
  const double nf = (double)n;
  red[tid] = ((double)t1 * (nf - 1.0) / nf + (double)t2 + (double)t3) / nf;
  __syncthreads();
  for (int s = 128; s > 0; s >>= 1) {
    if (tid < s) red[tid] += red[tid + s];
    __syncthreads();
  }
  if (tid == 0) partial[slice * NB + b] = red[0];  // every slot written; no init
}

__global__ __launch_bounds__(256) void pf_loss_finalize(
    const double* __restrict__ partial, float* __restrict__ out)
{
  __shared__ double red[256];
  const int tid = threadIdx.x;
  double s = 0.0;
#pragma unroll
  for (int k = 0; k < SPLIT; ++k) s += partial[k * NB + tid];  // fixed order
  red[tid] = s;
  __syncthreads();
  for (int st = 128; st > 0; st >>= 1) {
    if (tid < st) red[tid] += red[tid + st];
    __syncthreads();
  }
  if (tid == 0) out[0] = (float)(red[0] / (double)NB);
}

extern "C" void kernel_launch(void* const* d_in, const int* in_sizes, int n_in,
                              void* d_out, int out_size, void* d_ws,
                              size_t ws_size, hipStream_t stream) {
  (void)in_sizes; (void)n_in; (void)out_size; (void)ws_size;
  const float* pf      = (const float*)d_in[0]; // partial_forces [TOTAL,192,3]
  const float* pos     = (const float*)d_in[1]; // positions      [TOTAL,3]
  const int*   n_atoms = (const int*)d_in[2];   // [256]
  double*      part    = (double*)d_ws;         // SPLIT*256 doubles = 8 KB

  pf_loss_per_mol<<<dim3(NB, SPLIT), 256, 0, stream>>>(pf, pos, n_atoms, part);
  pf_loss_finalize<<<1, 256, 0, stream>>>(part, (float*)d_out);
}